// GeometricBilinear_13795434955476
// MI455X (gfx1250) — compile-verified
//
#include <hip/hip_runtime.h>

typedef __attribute__((ext_vector_type(2))) float v2f;
typedef __attribute__((ext_vector_type(8))) float v8f;

#define TOKS 32             // tokens per workgroup (two 16-column WMMA sub-tiles)
#define XROW 516            // 512 + 4 pad floats (bank-conflict avoidance)
#define LRPAD 17            // padded blade dim

// workspace float layout
#define WS_WL   0
#define WS_WR   131072
#define WS_WJL  262144
#define WS_WJR  393216
#define WS_WO   524288
#define WS_WEND 786432      // floats; sparse tables (ints) follow
#define TAB_STRIDE 544      // ints per table: 20 header + 256 entries * 2 dwords

__device__ __forceinline__ int swz_idx(int r, int c) {
  // fragment-swizzled index for V_WMMA_F32_16X16X4_F32 A-operand:
  // lane = kh*16 + (r&15), elements {col, col+1} contiguous per lane
  return ((((r >> 4) * 128 + (c >> 2)) * 32) + (((c >> 1) & 1) << 4) + (r & 15)) * 2 + (c & 1);
}

// ---------------- setup kernel 1: fuse w_mv with basis into swizzled GEMM weights
__global__ void fuse_weights_kernel(const float* __restrict__ basis,
                                    const float* __restrict__ wl,
                                    const float* __restrict__ wr,
                                    const float* __restrict__ wjl,
                                    const float* __restrict__ wjr,
                                    const float* __restrict__ wo,
                                    float* __restrict__ ws) {
  int gid = blockIdx.x * blockDim.x + threadIdx.x;
  if (gid < 524288) {                       // four 256x512 matrices
    int mat = gid >> 17;
    int rem = gid & 131071;
    int r = rem >> 9, c = rem & 511;
    int o = r >> 4, j = r & 15, i = c >> 4, k = c & 15;
    const float* wmv = (mat == 0) ? wl : (mat == 1) ? wr : (mat == 2) ? wjl : wjr;
    float v = 0.f;
#pragma unroll
    for (int b = 0; b < 9; ++b)
      v += wmv[(o * 32 + i) * 9 + b] * basis[b * 256 + j * 16 + k];
    ws[mat * 131072 + swz_idx(r, c)] = v;
  } else {                                   // 512x512 output matrix
    int rem = gid - 524288;
    int r = rem >> 9, c = rem & 511;
    int o = r >> 4, j = r & 15, cH = c >> 4, k = c & 15;
    float v = 0.f;
#pragma unroll
    for (int b = 0; b < 9; ++b)
      v += wo[(o * 32 + cH) * 9 + b] * basis[b * 256 + j * 16 + k];
    ws[WS_WO + swz_idx(r, c)] = v;
  }
}

// ---------------- setup kernel 2: compact gp / join tables into row-sorted sparse lists
__global__ void build_tables_kernel(const float* __restrict__ gp,
                                    const float* __restrict__ jn,
                                    int* __restrict__ tab) {
  int t = threadIdx.x;
  if (t >= 2) return;
  const float* T = (t == 0) ? gp : jn;
  int* row = tab + t * TAB_STRIDE;
  int* ent = row + 20;
  int cnt = 0;
  for (int i = 0; i < 16; ++i) {
    row[i] = cnt;
    for (int j = 0; j < 16; ++j)
      for (int k = 0; k < 16; ++k) {
        float s = T[(i * 16 + j) * 16 + k];
        if (s != 0.f && cnt < 256) {
          ent[cnt * 2]     = j | (k << 8);
          ent[cnt * 2 + 1] = __float_as_int(s);
          ++cnt;
        }
      }
  }
  row[16] = cnt;
}

// ---------------- WMMA helper: one A-tile row vs TWO token sub-tiles (A loaded once)
__device__ __forceinline__ void wmma_tile2(const float* __restrict__ Wbase, int mt,
                                           const float* __restrict__ Xs,
                                           int nn, int kh, int lane,
                                           v8f& acc0, v8f& acc1) {
  v8f a0 = {0.f, 0.f, 0.f, 0.f, 0.f, 0.f, 0.f, 0.f};
  v8f a1 = {0.f, 0.f, 0.f, 0.f, 0.f, 0.f, 0.f, 0.f};
  const v2f* Wf = (const v2f*)Wbase + mt * 128 * 32;
  const float* xb0 = Xs + nn * XROW + kh * 2;
  const float* xb1 = xb0 + 16 * XROW;
#pragma unroll 4
  for (int ks = 0; ks < 128; ++ks) {
    v2f a  = Wf[ks * 32 + lane];                      // coalesced b64, loaded ONCE per ks
    v2f b0 = *(const v2f*)(xb0 + ks * 4);             // ds_load b64, conflict-free
    v2f b1 = *(const v2f*)(xb1 + ks * 4);
    a0 = __builtin_amdgcn_wmma_f32_16x16x4_f32(false, a, false, b0, (short)0, a0, false, false);
    a1 = __builtin_amdgcn_wmma_f32_16x16x4_f32(false, a, false, b1, (short)0, a1, false, false);
  }
  acc0 = a0; acc1 = a1;
}

__device__ __forceinline__ void store_tile_lr(float* __restrict__ LRs, int msel, int mt,
                                              v8f acc, int tok, int kh) {
  float* dst = &LRs[((msel * TOKS + tok) * 16 + mt) * LRPAD + kh * 8];
#pragma unroll
  for (int v = 0; v < 8; ++v) dst[v] = acc[v];
}

__device__ __forceinline__ void bilinear(const int* __restrict__ tabT,
                                         const float* __restrict__ Lrow,
                                         const float* __restrict__ Rrow,
                                         float scale, float* __restrict__ res) {
  const int* row = tabT;
  const int* ent = tabT + 20;
#pragma unroll
  for (int i = 0; i < 16; ++i) {
    float a = 0.f;
    int e0 = row[i], e1 = row[i + 1];
    for (int e = e0; e < e1; ++e) {
      int jk  = ent[2 * e];
      float s = __int_as_float(ent[2 * e + 1]);
      a += s * Lrow[jk & 255] * Rrow[jk >> 8];
    }
    res[i] = a * scale;
  }
}

// ---------------- main fused kernel: 32 tokens per workgroup
__global__ void __launch_bounds__(256, 1)
geobilinear_kernel(const float* __restrict__ mv, const float* __restrict__ refmv,
                   const float* __restrict__ scl,
                   const float* __restrict__ wls,  const float* __restrict__ wrs,
                   const float* __restrict__ wjls, const float* __restrict__ wjrs,
                   const float* __restrict__ wo_s2mv,
                   const float* __restrict__ wo_mvs2s,
                   const float* __restrict__ wo_s2s,
                   const float* __restrict__ Wws, const int* __restrict__ tab,
                   float* __restrict__ outMV, float* __restrict__ outS) {
  __shared__ float Xs[TOKS * XROW];                    // X tile; later reused as hidden
  __shared__ float LRs[2 * TOKS * 16 * LRPAD];         // left/right then jl/jr
  __shared__ float Ss[TOKS * 64];
  __shared__ float Ps[TOKS];

  const int tid = threadIdx.x;
  const int wave = tid >> 5, lane = tid & 31;
  const int nn = lane & 15, kh = lane >> 4;
  const long tok0 = (long)blockIdx.x * TOKS;

  // ---- stage 0: stage inputs into LDS
  {
    const float4* mv4 = (const float4*)mv;
    for (int u = tid; u < TOKS * 128; u += 256) {
      int tok = u >> 7, g = u & 127;
      float4 v = mv4[(tok0 + tok) * 128 + g];
      *(float4*)&Xs[tok * XROW + g * 4] = v;
    }
    const float4* s4 = (const float4*)scl;
    for (int u = tid; u < TOKS * 16; u += 256) {
      int tok = u >> 4, g = u & 15;
      *(float4*)&Ss[tok * 64 + g * 4] = s4[(tok0 + tok) * 16 + g];
    }
    if (tid < TOKS) Ps[tid] = refmv[(tok0 + tid) * 16 + 15];
  }
  __syncthreads();

  const int bc = tid & 15;
  float gpReg[2][16], jnReg[2][16];

  // ---- stage 1a: left / right equi-linears (WMMA), 32 tile-rows over 8 waves
#pragma unroll
  for (int q = 0; q < 4; ++q) {
    int tile = q * 8 + wave;
    int msel = tile >> 4, mt = tile & 15;
    v8f acc0, acc1;
    wmma_tile2(Wws + (msel ? WS_WR : WS_WL), mt, Xs, nn, kh, lane, acc0, acc1);
    store_tile_lr(LRs, msel, mt, acc0, nn, kh);
    store_tile_lr(LRs, msel, mt, acc1, nn + 16, kh);
  }
  __syncthreads();
  for (int u = tid; u < 1024; u += 256) {              // scalar -> blade0
    int msel = u >> 9, tok = (u >> 4) & 31, c = u & 15;
    const float* wsc = msel ? wrs : wls;
    float d = 0.f;
#pragma unroll 8
    for (int s2 = 0; s2 < 64; ++s2) d += wsc[c * 64 + s2] * Ss[tok * 64 + s2];
    LRs[((msel * TOKS + tok) * 16 + c) * LRPAD] += d;
  }
  __syncthreads();

  // ---- stage 2a: sparse geometric product -> registers (2 tokens per thread)
#pragma unroll
  for (int p = 0; p < 2; ++p) {
    int tok = (tid >> 4) + 16 * p;
    bilinear(tab,
             &LRs[((0 * TOKS + tok) * 16 + bc) * LRPAD],
             &LRs[((1 * TOKS + tok) * 16 + bc) * LRPAD],
             1.f, gpReg[p]);
  }
  __syncthreads();

  // ---- stage 1b: jl / jr equi-linears (reuse LR buffer)
#pragma unroll
  for (int q = 0; q < 4; ++q) {
    int tile = q * 8 + wave;
    int msel = tile >> 4, mt = tile & 15;
    v8f acc0, acc1;
    wmma_tile2(Wws + (msel ? WS_WJR : WS_WJL), mt, Xs, nn, kh, lane, acc0, acc1);
    store_tile_lr(LRs, msel, mt, acc0, nn, kh);
    store_tile_lr(LRs, msel, mt, acc1, nn + 16, kh);
  }
  __syncthreads();
  for (int u = tid; u < 1024; u += 256) {
    int msel = u >> 9, tok = (u >> 4) & 31, c = u & 15;
    const float* wsc = msel ? wjrs : wjls;
    float d = 0.f;
#pragma unroll 8
    for (int s2 = 0; s2 < 64; ++s2) d += wsc[c * 64 + s2] * Ss[tok * 64 + s2];
    LRs[((msel * TOKS + tok) * 16 + c) * LRPAD] += d;
  }
  __syncthreads();

  // ---- stage 2b: sparse join (scaled by pseudoscalar of reference_mv)
#pragma unroll
  for (int p = 0; p < 2; ++p) {
    int tok = (tid >> 4) + 16 * p;
    bilinear(tab + TAB_STRIDE,
             &LRs[((0 * TOKS + tok) * 16 + bc) * LRPAD],
             &LRs[((1 * TOKS + tok) * 16 + bc) * LRPAD],
             Ps[tok], jnReg[p]);
  }
  __syncthreads();

  // ---- hidden overwrites dead X buffer, in GEMM-B layout
#pragma unroll
  for (int p = 0; p < 2; ++p) {
    int tok = (tid >> 4) + 16 * p;
    float* hrow = &Xs[tok * XROW];
#pragma unroll
    for (int i = 0; i < 16; ++i) hrow[bc * 16 + i] = gpReg[p][i];
#pragma unroll
    for (int i = 0; i < 16; ++i) hrow[(16 + bc) * 16 + i] = jnReg[p][i];
  }
  __syncthreads();

  // ---- stage 3: output equi-linear 512x512 (WMMA), write out_mv directly
#pragma unroll
  for (int q = 0; q < 4; ++q) {
    int mt = q * 8 + wave;                             // output channel o = mt
    v8f acc0, acc1;
    wmma_tile2(Wws + WS_WO, mt, Xs, nn, kh, lane, acc0, acc1);
    float e0 = 0.f, e1 = 0.f;
    if (kh == 0) {                                     // blade 0 lives in acc[0] of lanes 0-15
#pragma unroll 8
      for (int s2 = 0; s2 < 64; ++s2) {
        float w = wo_s2mv[mt * 64 + s2];
        e0 += w * Ss[nn * 64 + s2];
        e1 += w * Ss[(16 + nn) * 64 + s2];
      }
    }
    acc0[0] += e0;
    acc1[0] += e1;
    float* d0 = outMV + ((tok0 + nn) * 32 + mt) * 16 + kh * 8;
    float* d1 = outMV + ((tok0 + 16 + nn) * 32 + mt) * 16 + kh * 8;
    float4 f;
    f = (float4){acc0[0], acc0[1], acc0[2], acc0[3]}; *(float4*)d0 = f;
    f = (float4){acc0[4], acc0[5], acc0[6], acc0[7]}; *(float4*)(d0 + 4) = f;
    f = (float4){acc1[0], acc1[1], acc1[2], acc1[3]}; *(float4*)d1 = f;
    f = (float4){acc1[4], acc1[5], acc1[6], acc1[7]}; *(float4*)(d1 + 4) = f;
  }

  // ---- stage 4: scalar head
  for (int u = tid; u < TOKS * 64; u += 256) {
    int tok = u >> 6, o = u & 63;
    float a = 0.f;
#pragma unroll 8
    for (int c2 = 0; c2 < 32; ++c2) a += wo_mvs2s[o * 32 + c2] * Xs[tok * XROW + c2 * 16];
#pragma unroll 8
    for (int s2 = 0; s2 < 64; ++s2) a += wo_s2s[o * 64 + s2] * Ss[tok * 64 + s2];
    outS[(tok0 + tok) * 64 + o] = a;
  }
}

extern "C" void kernel_launch(void* const* d_in, const int* in_sizes, int n_in,
                              void* d_out, int out_size, void* d_ws, size_t ws_size,
                              hipStream_t stream) {
  (void)in_sizes; (void)n_in; (void)out_size; (void)ws_size;
  const float* mvp     = (const float*)d_in[0];
  const float* refp    = (const float*)d_in[1];
  const float* sclp    = (const float*)d_in[2];
  const float* basis   = (const float*)d_in[3];
  const float* gp      = (const float*)d_in[4];
  const float* jn      = (const float*)d_in[5];
  const float* wlmv    = (const float*)d_in[6];
  const float* wls     = (const float*)d_in[7];
  const float* wrmv    = (const float*)d_in[8];
  const float* wrs     = (const float*)d_in[9];
  const float* wjlmv   = (const float*)d_in[10];
  const float* wjls    = (const float*)d_in[11];
  const float* wjrmv   = (const float*)d_in[12];
  const float* wjrs    = (const float*)d_in[13];
  const float* womv    = (const float*)d_in[14];
  const float* wos2mv  = (const float*)d_in[15];
  const float* womvs2s = (const float*)d_in[16];
  const float* wos2s   = (const float*)d_in[17];

  float* wsF = (float*)d_ws;
  int* tab   = (int*)d_ws + WS_WEND;
  float* outMV = (float*)d_out;
  float* outS  = outMV + (size_t)32768 * 32 * 16;

  fuse_weights_kernel<<<3072, 256, 0, stream>>>(basis, wlmv, wrmv, wjlmv, wjrmv, womv, wsF);
  build_tables_kernel<<<1, 32, 0, stream>>>(gp, jn, tab);
  geobilinear_kernel<<<1024, 256, 0, stream>>>(mvp, refp, sclp, wls, wrs, wjls, wjrs,
                                               wos2mv, womvs2s, wos2s, wsF, tab,
                                               outMV, outS);
}